// FTDSM_54331336295084
// MI455X (gfx1250) — compile-verified
//
#include <hip/hip_runtime.h>
#include <math.h>

// ---------------- problem constants ----------------
#define BSZ   16
#define NW    17
#define NR    116
#define WS    10
#define EE    128
#define LNUM  5
#define NEXP  4
#define DS    16
#define SEQL  (NR*WS)        // 1160
#define EDIM  (NW*WS)        // 170
#define FB    (SEQL/2 + 1)   // 581  rfft bins
#define LAM   0.01f
#define EPSV  1e-5f
#define TWOPI 6.283185307179586f

#define GRPS  4              // m-tile groups per (b,w) slice
#define WPAD  132            // padded LDS row stride (132%64=4 -> conflict-free)

typedef __attribute__((ext_vector_type(2))) float v2f;
typedef __attribute__((ext_vector_type(8))) float v8f;
typedef __attribute__((ext_vector_type(4))) int   v4i;

typedef __attribute__((address_space(1))) v4i* gv4i_p;   // global int4*
typedef __attribute__((address_space(3))) v4i* lv4i_p;   // LDS int4*

#if __has_builtin(__builtin_amdgcn_global_load_async_to_lds_b128)
#define HAVE_ASYNC 1
#else
#define HAVE_ASYNC 0
#endif

// stage 32B (8 floats) global -> LDS; async on gfx1250 if available
__device__ __forceinline__ void stage32B(const float* g, float* l) {
#if HAVE_ASYNC
  __builtin_amdgcn_global_load_async_to_lds_b128(
      (gv4i_p)(g),     (lv4i_p)(l), 0, 0);
  __builtin_amdgcn_global_load_async_to_lds_b128(
      (gv4i_p)(g + 4), (lv4i_p)(l + 4), 0, 0);
#else
  float4 a = *(const float4*)g;
  float4 b = *(const float4*)(g + 4);
  *(float4*)l = a;
  *(float4*)(l + 4) = b;
#endif
}

__device__ __forceinline__ void wait_async_tiles(bool pending_next) {
#if HAVE_ASYNC
#if __has_builtin(__builtin_amdgcn_s_wait_asynccnt)
  if (pending_next) __builtin_amdgcn_s_wait_asynccnt(4);
  else              __builtin_amdgcn_s_wait_asynccnt(0);
#else
  if (pending_next) asm volatile("s_wait_asynccnt 0x4" ::: "memory");
  else              asm volatile("s_wait_asynccnt 0x0" ::: "memory");
#endif
#endif
}

// ================= small helpers =================
__device__ __forceinline__ float softplusf(float x) {
  return (x > 20.f) ? x : log1pf(expf(x));
}
__device__ __forceinline__ float sigmoidf_(float x) {
  return 1.f / (1.f + expf(-x));
}

// ======== softmax of w_c rows (NW x EE) ========
__global__ void k_wc_softmax(const float* __restrict__ w_c, float* __restrict__ wc) {
  int w = blockIdx.x * blockDim.x + threadIdx.x;
  if (w >= NW) return;
  const float* r = w_c + (size_t)w * EE;
  float m = r[0];
  for (int j = 1; j < EE; j++) m = fmaxf(m, r[j]);
  float s = 0.f;
  for (int j = 0; j < EE; j++) s += expf(r[j] - m);
  float inv = 1.f / s;
  for (int j = 0; j < EE; j++) wc[(size_t)w * EE + j] = expf(r[j] - m) * inv;
}

// ======== u = emb @ W_layer0 (per window, real & imag) ========
__global__ void k_uproj(const float* __restrict__ emb, const float* __restrict__ fgc_w,
                        float* __restrict__ ur, float* __restrict__ ui) {
  int idx = blockIdx.x * blockDim.x + threadIdx.x;
  if (idx >= NW * EE) return;
  int j = idx % EE, w = idx / EE;
  const float* wr0 = fgc_w + (size_t)(w * LNUM * 2) * EE * EE;       // fgc_w[w,0,0]
  const float* wi0 = wr0 + (size_t)EE * EE;                          // fgc_w[w,0,1]
  float ar = 0.f, ai = 0.f;
  for (int i = 0; i < EE; i++) {
    float e = emb[i];
    ar += e * wr0[(size_t)i * EE + j];
    ai += e * wi0[(size_t)i * EE + j];
  }
  ur[idx] = ar; ui[idx] = ai;
}

// ======== rfft (ortho) of win_seq rows: S[b,w,f] ========
__global__ __launch_bounds__(128) void k_dft(const float* __restrict__ seq,
                                             float* __restrict__ Sr, float* __restrict__ Si) {
  __shared__ float ss[SEQL];
  int w = blockIdx.y, b = blockIdx.z;
  const float* src = seq + ((size_t)(b * NW + w)) * SEQL;
  for (int i = threadIdx.x; i < SEQL; i += 128) ss[i] = src[i];
  __syncthreads();
  int f = blockIdx.x * 128 + threadIdx.x;
  if (f >= FB) return;
  float cr = 0.f, ci = 0.f;
  for (int t = 0; t < SEQL; t++) {
    int ft = (f * t) % SEQL;                 // keep phase small & exact
    float ang = (TWOPI / (float)SEQL) * (float)ft;
    float sn, cs; __sincosf(ang, &sn, &cs);
    float v = ss[t];
    cr += v * cs; ci -= v * sn;
  }
  float sc = rsqrtf((float)SEQL);            // norm='ortho'
  size_t o = ((size_t)(b * NW + w)) * FB + f;
  Sr[o] = cr * sc; Si[o] = ci * sc;
}

// ======== FGC layer 0 (rank-1 closed form) ========
__global__ void k_layer0(const float* __restrict__ Sr, const float* __restrict__ Si,
                         const float* __restrict__ ur, const float* __restrict__ ui,
                         const float* __restrict__ fgc_b,
                         float* __restrict__ Xr, float* __restrict__ Xi) {
  size_t idx = (size_t)blockIdx.x * blockDim.x + threadIdx.x;
  size_t total = (size_t)BSZ * NW * FB * EE;
  if (idx >= total) return;
  int j = (int)(idx & (EE - 1));
  size_t rest = idx >> 7;              // /EE
  int f = (int)(rest % FB);
  int bw = (int)(rest / FB);
  int w = bw % NW;
  float sr = Sr[(size_t)bw * FB + f], si = Si[(size_t)bw * FB + f];
  float uur = ur[(size_t)w * EE + j], uui = ui[(size_t)w * EE + j];
  const float* br = fgc_b + (size_t)(w * LNUM * 2) * EE;   // fgc_b[w,0,0]
  const float* bi = br + EE;
  float o_r = sr * uur - si * uui + br[j];
  float o_i = si * uur + sr * uui + bi[j];
  o_r = fmaxf(o_r, 0.f); o_r = (o_r > LAM) ? o_r - LAM : 0.f;   // softshrink(relu)
  o_i = fmaxf(o_i, 0.f); o_i = (o_i > LAM) ? o_i - LAM : 0.f;
  Xr[idx] = o_r; Xi[idx] = o_i;
}

// ======== FGC layers 1..4 : complex GEMM via V_WMMA_F32_16X16X4_F32 ========
// block = 256 (8 waves) owns one (b,w) slice's m-tile group:
//  - transposed weights staged once in LDS (both operands fed by DS loads)
//  - X tiles double-buffered via GLOBAL_LOAD_ASYNC_TO_LDS_B128 (if available)
// LDS: 2*128*132*4 + 2*2*16*132*4 = ~169 KB (one workgroup per WGP, 320KB limit)
__global__ __launch_bounds__(256) void k_fgc_layer(float* __restrict__ Xr,
                                                   float* __restrict__ Xi,
                                                   const float* __restrict__ fgc_w,
                                                   const float* __restrict__ fgc_b,
                                                   int layer) {
  __shared__ float sWrT[EE][WPAD];      // wr^T [n][k]
  __shared__ float sWiT[EE][WPAD];      // wi^T [n][k]
  __shared__ float sXr[2][16][WPAD];    // double-buffered tile
  __shared__ float sXi[2][16][WPAD];
  int grp = blockIdx.x;                 // 0..GRPS-1
  int w = blockIdx.y, b = blockIdx.z;
  int tid = threadIdx.x, lane = tid & 31, wave = tid >> 5;
  size_t xbase = ((size_t)(b * NW + w)) * FB * EE;

  const float* wr = fgc_w + (size_t)((w * LNUM + layer) * 2) * EE * EE;
  const float* wi = wr + (size_t)EE * EE;
  const float* br = fgc_b + (size_t)((w * LNUM + layer) * 2) * EE;
  const float* bi = br + EE;

  // ---- stage transposed weights into LDS (once per block) ----
  for (int i = tid * 4; i < EE * EE; i += 256 * 4) {
    int k = i >> 7;            // source row  (K)
    int n = i & (EE - 1);      // source col  (N), 4 consecutive
    float4 vr = *(const float4*)(wr + i);
    float4 vi = *(const float4*)(wi + i);
    sWrT[n + 0][k] = vr.x; sWrT[n + 1][k] = vr.y; sWrT[n + 2][k] = vr.z; sWrT[n + 3][k] = vr.w;
    sWiT[n + 0][k] = vi.x; sWiT[n + 1][k] = vi.y; sWiT[n + 2][k] = vi.z; sWiT[n + 3][k] = vi.w;
  }

  // ---- m-tile list for this block ----
  const int NMT = (FB + 15) / 16;       // 37
  int tiles[(NMT + GRPS - 1) / GRPS];
  int ntiles = 0;
  for (int mt = grp; mt < NMT; mt += GRPS) tiles[ntiles++] = mt;

  // per-thread staging slot: one 32B chunk of one row, per component
  int srow = tid >> 4;                  // 0..15
  int scol = (tid & 15) * 8;            // 0,8,...,120

  // prefetch first tile into buffer 0
  {
    int mt = tiles[0];
    stage32B(Xr + xbase + ((size_t)mt * 16 + srow) * EE + scol, &sXr[0][srow][scol]);
    stage32B(Xi + xbase + ((size_t)mt * 16 + srow) * EE + scol, &sXi[0][srow][scol]);
  }

  int mrow  = lane & 15;                // A-row within tile (16x4 f32 layout)
  int khalf = (lane >> 4) * 2;          // lanes 16-31 hold K=2,3
  int ncol  = wave * 16 + (lane & 15);
  float bre = br[ncol], bie = bi[ncol];

  for (int ti = 0; ti < ntiles; ti++) {
    int cur = ti & 1;
    bool more = (ti + 1 < ntiles);
    if (more) {
      int mt = tiles[ti + 1];
      stage32B(Xr + xbase + ((size_t)mt * 16 + srow) * EE + scol, &sXr[cur ^ 1][srow][scol]);
      stage32B(Xi + xbase + ((size_t)mt * 16 + srow) * EE + scol, &sXi[cur ^ 1][srow][scol]);
    }
    wait_async_tiles(more);
    __syncthreads();                    // tile (and, first time, weights) visible

    v8f accr = {}; v8f acci = {};
    for (int k = 0; k < EE; k += 4) {
      v2f axr, axi, bwr, bwi;
      axr.x = sXr[cur][mrow][k + khalf];  axr.y = sXr[cur][mrow][k + khalf + 1];
      axi.x = sXi[cur][mrow][k + khalf];  axi.y = sXi[cur][mrow][k + khalf + 1];
      v2f axin = -axi;                    // f32 WMMA has no A/B neg modifier
      bwr.x = sWrT[ncol][k + khalf];      bwr.y = sWrT[ncol][k + khalf + 1];
      bwi.x = sWiT[ncol][k + khalf];      bwi.y = sWiT[ncol][k + khalf + 1];
      accr = __builtin_amdgcn_wmma_f32_16x16x4_f32(false, axr,  false, bwr, (short)0, accr, false, false);
      accr = __builtin_amdgcn_wmma_f32_16x16x4_f32(false, axin, false, bwi, (short)0, accr, false, false);
      acci = __builtin_amdgcn_wmma_f32_16x16x4_f32(false, axi,  false, bwr, (short)0, acci, false, false);
      acci = __builtin_amdgcn_wmma_f32_16x16x4_f32(false, axr,  false, bwi, (short)0, acci, false, false);
    }

    // epilogue: C/D layout -> vgpr r holds M = r + (lane>=16 ? 8 : 0)
    int mtile = tiles[ti];
    int rbase = (lane >> 4) * 8;
    for (int r = 0; r < 8; r++) {
      int grow = mtile * 16 + rbase + r;
      if (grow < FB) {
        float o_r = accr[r] + bre; o_r = fmaxf(o_r, 0.f); o_r = (o_r > LAM) ? o_r - LAM : 0.f;
        float o_i = acci[r] + bie; o_i = fmaxf(o_i, 0.f); o_i = (o_i > LAM) ? o_i - LAM : 0.f;
        Xr[xbase + (size_t)grow * EE + ncol] = o_r;
        Xi[xbase + (size_t)grow * EE + ncol] = o_i;
      }
    }
    __syncthreads();                    // all reads of buf done before its reuse
  }
}

// ======== contract Xf with softmax(w_c) over E (in freq domain) ========
__global__ void k_wc_contract(const float* __restrict__ Xr, const float* __restrict__ Xi,
                              const float* __restrict__ wc,
                              float* __restrict__ Zr, float* __restrict__ Zi) {
  int idx = blockIdx.x * blockDim.x + threadIdx.x;
  if (idx >= BSZ * NW * FB) return;
  int bw = idx / FB;
  int w = bw % NW;
  const float* xr = Xr + (size_t)idx * EE;
  const float* xi = Xi + (size_t)idx * EE;
  const float* wv = wc + (size_t)w * EE;
  float ar = 0.f, ai = 0.f;
  for (int e = 0; e < EE; e++) { ar += xr[e] * wv[e]; ai += xi[e] * wv[e]; }
  Zr[idx] = ar; Zi[idx] = ai;
}

// ======== irfft (ortho) of Z -> freq_feat ========
__global__ __launch_bounds__(256) void k_idft(const float* __restrict__ Zr,
                                              const float* __restrict__ Zi,
                                              float* __restrict__ ff) {
  __shared__ float sZr[FB];
  __shared__ float sZi[FB];
  int w = blockIdx.y, b = blockIdx.z;
  size_t zb = ((size_t)(b * NW + w)) * FB;
  for (int i = threadIdx.x; i < FB; i += 256) { sZr[i] = Zr[zb + i]; sZi[i] = Zi[zb + i]; }
  __syncthreads();
  int t = blockIdx.x * 256 + threadIdx.x;
  if (t >= SEQL) return;
  float acc = sZr[0];                              // DC (imag ignored, C2R convention)
  for (int f = 1; f < FB - 1; f++) {
    int ft = (f * t) % SEQL;
    float ang = (TWOPI / (float)SEQL) * (float)ft;
    float sn, cs; __sincosf(ang, &sn, &cs);
    acc += 2.f * (sZr[f] * cs - sZi[f] * sn);
  }
  acc += sZr[FB - 1] * ((t & 1) ? -1.f : 1.f);     // Nyquist
  ff[((size_t)(b * NW + w)) * SEQL + t] = acc * rsqrtf((float)SEQL);
}

// ======== time branch: x1 = mean over windows of win_pcc ========
__global__ void k_x1mean(const float* __restrict__ pcc, float* __restrict__ x1) {
  int idx = blockIdx.x * blockDim.x + threadIdx.x;
  if (idx >= BSZ * NR * NR) return;
  const float* p = pcc + (size_t)idx * NW;
  float a = 0.f;
  for (int w = 0; w < NW; w++) a += p[w];
  x1[idx] = a * (1.f / NW);
}

// dt1 = softplus(x1 @ Wdt1)
__global__ void k_dt1(const float* __restrict__ x1, const float* __restrict__ Wdt1,
                      float* __restrict__ dt1) {
  int idx = blockIdx.x * blockDim.x + threadIdx.x;
  if (idx >= BSZ * NR * NR) return;
  int j = idx % NR; int bi = idx / NR;     // bi = b*NR + tt
  const float* row = x1 + (size_t)bi * NR;
  float a = 0.f;
  for (int m = 0; m < NR; m++) a += row[m] * Wdt1[(size_t)m * NR + j];
  dt1[idx] = softplusf(a);
}

// Bt1/Ct1 = x1 @ WB1 / WC1
__global__ void k_bc1(const float* __restrict__ x1, const float* __restrict__ WB1,
                      const float* __restrict__ WC1, float* __restrict__ Bt1,
                      float* __restrict__ Ct1) {
  int idx = blockIdx.x * blockDim.x + threadIdx.x;
  if (idx >= BSZ * NR * DS) return;
  int d = idx % DS; int bi = idx / DS;
  const float* row = x1 + (size_t)bi * NR;
  float ab = 0.f, ac = 0.f;
  for (int m = 0; m < NR; m++) {
    float v = row[m];
    ab += v * WB1[(size_t)m * DS + d];
    ac += v * WC1[(size_t)m * DS + d];
  }
  Bt1[idx] = ab; Ct1[idx] = ac;
}

// scan1 + pooled mean (y1 never materialized)
__global__ void k_scan1(const float* __restrict__ x1, const float* __restrict__ dt1,
                        const float* __restrict__ Bt1, const float* __restrict__ Ct1,
                        const float* __restrict__ A1, const float* __restrict__ D1,
                        float* __restrict__ pooled) {
  int idx = blockIdx.x * blockDim.x + threadIdx.x;
  if (idx >= BSZ * NR) return;
  int n = idx % NR, b = idx / NR;
  float h[DS];
  for (int s = 0; s < DS; s++) h[s] = 0.f;
  float psum = 0.f;
  for (int tt = 0; tt < NR; tt++) {
    size_t bt = (size_t)(b * NR + tt);
    float xt = x1[bt * NR + n];
    float dtt = dt1[bt * NR + n];
    float y = 0.f;
    for (int s = 0; s < DS; s++) {
      float dA = expf(dtt * A1[(size_t)n * DS + s]);
      h[s] = dA * h[s] + dtt * xt * Bt1[bt * DS + s];
      y += h[s] * Ct1[bt * DS + s];
    }
    psum += y + D1[n] * xt;
  }
  pooled[idx] = psum * (1.f / NR);
}

// out1 = einsum('bd,wdk->bwk', pooled, Wout1)
__global__ void k_out1(const float* __restrict__ pooled, const float* __restrict__ Wout1,
                       float* __restrict__ out1) {
  int idx = blockIdx.x * blockDim.x + threadIdx.x;
  if (idx >= BSZ * NW * SEQL) return;
  int k = idx % SEQL; int bw = idx / SEQL;
  int w = bw % NW, b = bw / NW;
  float a = 0.f;
  for (int d = 0; d < NR; d++)
    a += pooled[(size_t)b * NR + d] * Wout1[((size_t)w * NR + d) * SEQL + k];
  out1[idx] = a;
}

// x2 = einsum('bnm,bwmf->bwnf', global_adjs, freq_feat)
__global__ void k_x2(const float* __restrict__ gadj, const float* __restrict__ ff,
                     float* __restrict__ x2) {
  int idx = blockIdx.x * blockDim.x + threadIdx.x;
  if (idx >= BSZ * NW * NR * WS) return;
  int f = idx % WS; int rest = idx / WS;
  int n = rest % NR; rest /= NR;
  int w = rest % NW; int b = rest / NW;
  const float* grow = gadj + ((size_t)b * NR + n) * NR;
  const float* fr = ff + ((size_t)(b * NW + w)) * SEQL;
  float a = 0.f;
  for (int m = 0; m < NR; m++) a += grow[m] * fr[m * WS + f];
  x2[idx] = a;
}

// dt2 = softplus(x2 @ Wdt2)
__global__ void k_dt2(const float* __restrict__ x2, const float* __restrict__ Wdt2,
                      float* __restrict__ dt2) {
  int idx = blockIdx.x * blockDim.x + threadIdx.x;
  if (idx >= BSZ * NW * SEQL) return;
  int j = idx % SEQL; int bi = idx / SEQL;
  const float* row = x2 + (size_t)bi * SEQL;
  float a = 0.f;
  for (int m = 0; m < SEQL; m++) a += row[m] * Wdt2[(size_t)m * SEQL + j];
  dt2[idx] = softplusf(a);
}

__global__ void k_bc2(const float* __restrict__ x2, const float* __restrict__ WB2,
                      const float* __restrict__ WC2, float* __restrict__ Bt2,
                      float* __restrict__ Ct2) {
  int idx = blockIdx.x * blockDim.x + threadIdx.x;
  if (idx >= BSZ * NW * DS) return;
  int d = idx % DS; int bi = idx / DS;
  const float* row = x2 + (size_t)bi * SEQL;
  float ab = 0.f, ac = 0.f;
  for (int m = 0; m < SEQL; m++) {
    float v = row[m];
    ab += v * WB2[(size_t)m * DS + d];
    ac += v * WC2[(size_t)m * DS + d];
  }
  Bt2[idx] = ab; Ct2[idx] = ac;
}

__global__ void k_scan2(const float* __restrict__ x2, const float* __restrict__ dt2,
                        const float* __restrict__ Bt2, const float* __restrict__ Ct2,
                        const float* __restrict__ A2, const float* __restrict__ D2,
                        float* __restrict__ y2) {
  int idx = blockIdx.x * blockDim.x + threadIdx.x;
  if (idx >= BSZ * SEQL) return;
  int j = idx % SEQL, b = idx / SEQL;
  float h[DS];
  for (int s = 0; s < DS; s++) h[s] = 0.f;
  for (int tt = 0; tt < NW; tt++) {
    size_t bt = (size_t)(b * NW + tt);
    float xt = x2[bt * SEQL + j];
    float dtt = dt2[bt * SEQL + j];
    float y = 0.f;
    for (int s = 0; s < DS; s++) {
      float dA = expf(dtt * A2[(size_t)j * DS + s]);
      h[s] = dA * h[s] + dtt * xt * Bt2[bt * DS + s];
      y += h[s] * Ct2[bt * DS + s];
    }
    y2[bt * SEQL + j] = y + D2[j] * xt;
  }
}

// out2 = y2 @ Wout2
__global__ void k_out2(const float* __restrict__ y2, const float* __restrict__ Wout2,
                       float* __restrict__ out2) {
  int idx = blockIdx.x * blockDim.x + threadIdx.x;
  if (idx >= BSZ * NW * SEQL) return;
  int k = idx % SEQL; int bi = idx / SEQL;
  const float* row = y2 + (size_t)bi * SEQL;
  float a = 0.f;
  for (int m = 0; m < SEQL; m++) a += row[m] * Wout2[(size_t)m * SEQL + k];
  out2[idx] = a;
}

// layernorm over last dim SEQL; one block per row
__global__ __launch_bounds__(256) void k_ln(const float* __restrict__ in,
                                            const float* __restrict__ g,
                                            const float* __restrict__ bta,
                                            float* __restrict__ out) {
  __shared__ float s1[256], s2[256];
  int row = blockIdx.x, tid = threadIdx.x;
  const float* r = in + (size_t)row * SEQL;
  float a = 0.f, b2 = 0.f;
  for (int k = tid; k < SEQL; k += 256) { float v = r[k]; a += v; b2 += v * v; }
  s1[tid] = a; s2[tid] = b2; __syncthreads();
  for (int st = 128; st > 0; st >>= 1) {
    if (tid < st) { s1[tid] += s1[tid + st]; s2[tid] += s2[tid + st]; }
    __syncthreads();
  }
  float m = s1[0] / SEQL;
  float var = s2[0] / SEQL - m * m;
  float inv = rsqrtf(var + EPSV);
  for (int k = tid; k < SEQL; k += 256)
    out[(size_t)row * SEQL + k] = (r[k] - m) * inv * g[k] + bta[k];
}

// gate + fusion (cat layout: (b,n,w,[freq2 0..9, time 0..9]))
__global__ void k_fusion(const float* __restrict__ f2, const float* __restrict__ tf,
                         const float* __restrict__ Wg, const float* __restrict__ bg,
                         float* __restrict__ fus) {
  int idx = blockIdx.x * blockDim.x + threadIdx.x;
  if (idx >= BSZ * NR * EDIM) return;
  int g = idx % EDIM; int rest = idx / EDIM;
  int n = rest % NR, b = rest / NR;
  float acc = 0.f;
  for (int i = 0; i < 2 * EDIM; i++) {
    int w = i / 20, c = i % 20;
    const float* src = (c < 10) ? f2 : tf;
    float v = src[((size_t)(b * NW + w)) * SEQL + n * WS + (c % 10)];
    acc += v * Wg[(size_t)i * EDIM + g];
  }
  float gate = sigmoidf_(acc + bg[g]);
  int w2 = g / WS, t2 = g % WS;
  float fp = f2[((size_t)(b * NW + w2)) * SEQL + n * WS + t2];
  float tp = tf[((size_t)(b * NW + w2)) * SEQL + n * WS + t2];
  fus[idx] = gate * fp + (1.f - gate) * tp;
}

// router: LN(fusion) @ Wr + br -> top-2 softmax -> dense gates
__global__ void k_router(const float* __restrict__ fus, const float* __restrict__ rg,
                         const float* __restrict__ rb, const float* __restrict__ Wr,
                         const float* __restrict__ br, float* __restrict__ gts) {
  int idx = blockIdx.x * blockDim.x + threadIdx.x;
  if (idx >= BSZ * NR) return;
  const float* row = fus + (size_t)idx * EDIM;
  float s = 0.f, s2 = 0.f;
  for (int e = 0; e < EDIM; e++) { float v = row[e]; s += v; s2 += v * v; }
  float m = s / EDIM;
  float var = s2 / EDIM - m * m;
  float inv = rsqrtf(var + EPSV);
  float lg[NEXP];
  for (int k = 0; k < NEXP; k++) lg[k] = br[k];
  for (int e = 0; e < EDIM; e++) {
    float nv = (row[e] - m) * inv * rg[e] + rb[e];
    for (int k = 0; k < NEXP; k++) lg[k] += nv * Wr[(size_t)e * NEXP + k];
  }
  int i0 = 0; float v0 = lg[0];
  for (int k = 1; k < NEXP; k++) if (lg[k] > v0) { v0 = lg[k]; i0 = k; }
  int i1 = -1; float v1 = -3.4e38f;
  for (int k = 0; k < NEXP; k++) if (k != i0 && lg[k] > v1) { v1 = lg[k]; i1 = k; }
  float w0 = 1.f / (1.f + expf(v1 - v0));
  float w1 = 1.f - w0;
  float* go = gts + (size_t)idx * NEXP;
  for (int k = 0; k < NEXP; k++) go[k] = 0.f;
  go[i0] = w0; go[i1] = w1;
}

// support = einsum('bnf,efg->beng', fusion, We)
__global__ void k_support(const float* __restrict__ fus, const float* __restrict__ We,
                          float* __restrict__ sup) {
  int idx = blockIdx.x * blockDim.x + threadIdx.x;
  if (idx >= BSZ * NEXP * NR * EDIM) return;
  int g = idx % EDIM; int rest = idx / EDIM;
  int n = rest % NR; rest /= NR;
  int e = rest % NEXP; int b = rest / NEXP;
  const float* frow = fus + ((size_t)b * NR + n) * EDIM;
  const float* wbase = We + (size_t)e * EDIM * EDIM;
  float a = 0.f;
  for (int f = 0; f < EDIM; f++) a += frow[f] * wbase[(size_t)f * EDIM + g];
  sup[idx] = a;
}

// moe = sum_e gates * relu(G @ support)
__global__ void k_moe(const float* __restrict__ gadj, const float* __restrict__ sup,
                      const float* __restrict__ gts, float* __restrict__ moe) {
  int idx = blockIdx.x * blockDim.x + threadIdx.x;
  if (idx >= BSZ * NR * EDIM) return;
  int g = idx % EDIM; int rest = idx / EDIM;
  int n = rest % NR, b = rest / NR;
  const float* grow = gadj + ((size_t)b * NR + n) * NR;
  float acc = 0.f;
  for (int e = 0; e < NEXP; e++) {
    float gt = gts[((size_t)b * NR + n) * NEXP + e];
    float a = 0.f;
    const float* sbase = sup + ((size_t)(b * NEXP + e) * NR) * EDIM + g;
    for (int m = 0; m < NR; m++) a += grow[m] * sbase[(size_t)m * EDIM];
    acc += gt * fmaxf(a, 0.f);
  }
  moe[idx] = acc;
}

// generic FC: out = act((in@W + b) * scale)
__global__ void k_fc(const float* __restrict__ in, const float* __restrict__ W,
                     const float* __restrict__ bias, float* __restrict__ out,
                     int rows, int id, int od, int do_relu, float scale) {
  int idx = blockIdx.x * blockDim.x + threadIdx.x;
  if (idx >= rows * od) return;
  int o = idx % od, r = idx / od;
  const float* row = in + (size_t)r * id;
  float a = 0.f;
  for (int i = 0; i < id; i++) a += row[i] * W[(size_t)i * od + o];
  a = (a + bias[o]) * scale;
  if (do_relu) a = fmaxf(a, 0.f);
  out[idx] = a;
}

// ================= host launcher =================
extern "C" void kernel_launch(void* const* d_in, const int* in_sizes, int n_in,
                              void* d_out, int out_size, void* d_ws, size_t ws_size,
                              hipStream_t stream) {
  (void)in_sizes; (void)n_in; (void)out_size; (void)ws_size;
  const float* win_seq = (const float*)d_in[0];
  const float* win_pcc = (const float*)d_in[1];
  const float* gadj    = (const float*)d_in[2];
  const float* emb     = (const float*)d_in[4];
  const float* fgc_w   = (const float*)d_in[5];
  const float* fgc_b   = (const float*)d_in[6];
  const float* w_c     = (const float*)d_in[7];
  const float* ln_g    = (const float*)d_in[8];
  const float* ln_b    = (const float*)d_in[9];
  const float* A1      = (const float*)d_in[10];
  const float* WB1     = (const float*)d_in[11];
  const float* WC1     = (const float*)d_in[12];
  const float* Wdt1    = (const float*)d_in[13];
  const float* D1      = (const float*)d_in[14];
  const float* Wout1   = (const float*)d_in[15];
  const float* A2      = (const float*)d_in[16];
  const float* WB2     = (const float*)d_in[17];
  const float* WC2     = (const float*)d_in[18];
  const float* Wdt2    = (const float*)d_in[19];
  const float* D2      = (const float*)d_in[20];
  const float* Wout2   = (const float*)d_in[21];
  const float* W_gate  = (const float*)d_in[22];
  const float* b_gate  = (const float*)d_in[23];
  const float* rln_g   = (const float*)d_in[24];
  const float* rln_b   = (const float*)d_in[25];
  const float* Wr      = (const float*)d_in[26];
  const float* brr     = (const float*)d_in[27];
  const float* We      = (const float*)d_in[28];
  const float* W1      = (const float*)d_in[29];
  const float* b1      = (const float*)d_in[30];
  const float* W2      = (const float*)d_in[31];
  const float* b2      = (const float*)d_in[32];
  const float* W3      = (const float*)d_in[33];
  const float* b3      = (const float*)d_in[34];
  const float* W4      = (const float*)d_in[35];
  const float* b4      = (const float*)d_in[36];

  // -------- workspace layout (floats, ~182 MB total) --------
  float* ws = (float*)d_ws;
  size_t o = 0;
  float* Sr  = ws + o; o += (size_t)BSZ * NW * FB;
  float* Si  = ws + o; o += (size_t)BSZ * NW * FB;
  float* ur  = ws + o; o += (size_t)NW * EE;
  float* ui  = ws + o; o += (size_t)NW * EE;
  float* wc  = ws + o; o += (size_t)NW * EE;
  float* Xr  = ws + o; o += (size_t)BSZ * NW * FB * EE;
  float* Xi  = ws + o; o += (size_t)BSZ * NW * FB * EE;
  float* Zr  = ws + o; o += (size_t)BSZ * NW * FB;
  float* Zi  = ws + o; o += (size_t)BSZ * NW * FB;
  float* ff  = ws + o; o += (size_t)BSZ * NW * SEQL;
  float* x1  = ws + o; o += (size_t)BSZ * NR * NR;
  float* dt1 = ws + o; o += (size_t)BSZ * NR * NR;
  float* Bt1 = ws + o; o += (size_t)BSZ * NR * DS;
  float* Ct1 = ws + o; o += (size_t)BSZ * NR * DS;
  float* pooled = ws + o; o += (size_t)BSZ * NR;
  float* out1 = ws + o; o += (size_t)BSZ * NW * SEQL;
  float* x2   = ws + o; o += (size_t)BSZ * NW * SEQL;
  float* dt2  = ws + o; o += (size_t)BSZ * NW * SEQL;
  float* Bt2  = ws + o; o += (size_t)BSZ * NW * DS;
  float* Ct2  = ws + o; o += (size_t)BSZ * NW * DS;
  float* y2   = ws + o; o += (size_t)BSZ * NW * SEQL;
  float* out2 = ws + o; o += (size_t)BSZ * NW * SEQL;
  float* tf   = ws + o; o += (size_t)BSZ * NW * SEQL;
  float* f2   = ws + o; o += (size_t)BSZ * NW * SEQL;
  float* fus  = ws + o; o += (size_t)BSZ * NR * EDIM;
  float* gts  = ws + o; o += (size_t)BSZ * NR * NEXP;
  float* sup  = ws + o; o += (size_t)BSZ * NEXP * NR * EDIM;
  float* moe  = ws + o; o += (size_t)BSZ * NR * EDIM;
  float* h1   = ws + o; o += (size_t)BSZ * 512;
  float* h2   = ws + o; o += (size_t)BSZ * 256;
  float* h3   = ws + o; o += (size_t)BSZ * 128;

  const float bn_scale = 1.0f / sqrtf(1.0f + EPSV);
  auto nb = [](size_t n, int t) { return (unsigned)((n + t - 1) / t); };

  // ---- frequency branch ----
  k_wc_softmax<<<1, 32, 0, stream>>>(w_c, wc);
  k_uproj<<<nb(NW * EE, 256), 256, 0, stream>>>(emb, fgc_w, ur, ui);
  k_dft<<<dim3(nb(FB, 128), NW, BSZ), 128, 0, stream>>>(win_seq, Sr, Si);
  k_layer0<<<nb((size_t)BSZ * NW * FB * EE, 256), 256, 0, stream>>>(Sr, Si, ur, ui, fgc_b, Xr, Xi);
  for (int l = 1; l < LNUM; l++)
    k_fgc_layer<<<dim3(GRPS, NW, BSZ), 256, 0, stream>>>(Xr, Xi, fgc_w, fgc_b, l);
  k_wc_contract<<<nb((size_t)BSZ * NW * FB, 256), 256, 0, stream>>>(Xr, Xi, wc, Zr, Zi);
  k_idft<<<dim3(nb(SEQL, 256), NW, BSZ), 256, 0, stream>>>(Zr, Zi, ff);

  // ---- time branch (scan 1) ----
  k_x1mean<<<nb((size_t)BSZ * NR * NR, 256), 256, 0, stream>>>(win_pcc, x1);
  k_dt1<<<nb((size_t)BSZ * NR * NR, 256), 256, 0, stream>>>(x1, Wdt1, dt1);
  k_bc1<<<nb((size_t)BSZ * NR * DS, 256), 256, 0, stream>>>(x1, WB1, WC1, Bt1, Ct1);
  k_scan1<<<nb((size_t)BSZ * NR, 256), 256, 0, stream>>>(x1, dt1, Bt1, Ct1, A1, D1, pooled);
  k_out1<<<nb((size_t)BSZ * NW * SEQL, 256), 256, 0, stream>>>(pooled, Wout1, out1);

  // ---- freq branch (scan 2) ----
  k_x2<<<nb((size_t)BSZ * NW * NR * WS, 256), 256, 0, stream>>>(gadj, ff, x2);
  k_dt2<<<nb((size_t)BSZ * NW * SEQL, 256), 256, 0, stream>>>(x2, Wdt2, dt2);
  k_bc2<<<nb((size_t)BSZ * NW * DS, 256), 256, 0, stream>>>(x2, WB2, WC2, Bt2, Ct2);
  k_scan2<<<nb((size_t)BSZ * SEQL, 256), 256, 0, stream>>>(x2, dt2, Bt2, Ct2, A2, D2, y2);
  k_out2<<<nb((size_t)BSZ * NW * SEQL, 256), 256, 0, stream>>>(y2, Wout2, out2);

  // ---- fusion ----
  k_ln<<<BSZ * NW, 256, 0, stream>>>(out1, ln_g, ln_b, tf);
  k_ln<<<BSZ * NW, 256, 0, stream>>>(out2, ln_g, ln_b, f2);
  k_fusion<<<nb((size_t)BSZ * NR * EDIM, 256), 256, 0, stream>>>(f2, tf, W_gate, b_gate, fus);
  k_router<<<nb((size_t)BSZ * NR, 256), 256, 0, stream>>>(fus, rln_g, rln_b, Wr, brr, gts);
  k_support<<<nb((size_t)BSZ * NEXP * NR * EDIM, 256), 256, 0, stream>>>(fus, We, sup);
  k_moe<<<nb((size_t)BSZ * NR * EDIM, 256), 256, 0, stream>>>(gadj, sup, gts, moe);

  // ---- MLP head ----
  k_fc<<<nb((size_t)BSZ * 512, 256), 256, 0, stream>>>(moe, W1, b1, h1, BSZ, NR * EDIM, 512, 1, bn_scale);
  k_fc<<<nb((size_t)BSZ * 256, 256), 256, 0, stream>>>(h1, W2, b2, h2, BSZ, 512, 256, 1, bn_scale);
  k_fc<<<nb((size_t)BSZ * 128, 256), 256, 0, stream>>>(h2, W3, b3, h3, BSZ, 256, 128, 1, bn_scale);
  k_fc<<<nb((size_t)BSZ * 2, 256), 256, 0, stream>>>(h3, W4, b4, (float*)d_out, BSZ, 128, 2, 0, 1.0f);
}